// ShiftLSA_20392504721878
// MI455X (gfx1250) — compile-verified
//
#include <hip/hip_runtime.h>
#include <hip/hip_bf16.h>

// ---------------------------------------------------------------------------
// ShiftLSA on MI455X (gfx1250, wave32).
//
// corr[b,n,m] = (1/sqrt(128)) * sum_l attn[b,n,l] * G[b, p(n,l), p(m,l)]
//   G[b,u,v]  = sum_d F1t[b,u,d] * F2t[b,v,d]   (K=128 GEMM -> fp32 WMMA)
// Factorization cuts matrix FLOPs 25x vs the naive bhndl,bhmdl contraction.
// G (134 MB, both batches) fits the 192 MB L2, so stage C's 25x re-read
// (3.4 GB) stays on-chip.
// ---------------------------------------------------------------------------

typedef __attribute__((ext_vector_type(2))) float v2f;
typedef __attribute__((ext_vector_type(8))) float v8f;

#define NPIX 4096   // 64*64
#define CDIM 128
#define LWIN 25

__device__ __forceinline__ int clamp6(int v) {
    return v < 0 ? 0 : (v > 63 ? 63 : v);
}

// ---------------------------------------------------------------------------
// Stage A: F1t[b][pix][d] = sum_c W1[d][c] * fmap[b][c][pix]   (and F2t).
// Pixel-major layout gives contiguous K=128 rows for the WMMA GEMM.
// grid = 4096 blocks x 256 thr (2 pixels x 128 channels per block).
// ---------------------------------------------------------------------------
__global__ void __launch_bounds__(256)
shiftlsa_stageA(const float* __restrict__ fmap, const float* __restrict__ fmap2,
                const float* __restrict__ W1,   const float* __restrict__ W2,
                float* __restrict__ F1t, float* __restrict__ F2t) {
    const int t = threadIdx.x;
    const int d = t & 127;
    const int p = t >> 7;                       // 0..1
    const int P = blockIdx.x * 2 + p;           // 0..8191 = b*4096 + pix
    const int b = P >> 12;
    const int pix = P & 4095;

    const float* fm1 = fmap  + ((size_t)b * CDIM) * NPIX + pix;
    const float* fm2 = fmap2 + ((size_t)b * CDIM) * NPIX + pix;
    const float* w1  = W1 + (size_t)d * CDIM;
    const float* w2  = W2 + (size_t)d * CDIM;

    float s1 = 0.f, s2 = 0.f;
#pragma unroll 8
    for (int c = 0; c < CDIM; ++c) {
        const float u = fm1[(size_t)c * NPIX];   // broadcast across 128 lanes
        const float v = fm2[(size_t)c * NPIX];
        s1 = fmaf(w1[c], u, s1);
        s2 = fmaf(w2[c], v, s2);
    }
    F1t[(size_t)P * CDIM + d] = s1;
    F2t[(size_t)P * CDIM + d] = s2;
}

// ---------------------------------------------------------------------------
// Stage B: G[b] = F1t[b] (4096x128) x F2t[b]^T (128x4096), fp32 WMMA 16x16x4.
// Block = 256 thr (8 wave32), computes a 64x64 tile of G.
// LDS tiles stored [row][68] (68-dword stride => 4-bank rotation per row,
// conflict-free ds_load_b64 fragment fetches).
// Wave w: tile-row (w&3), tile-col pair (w>>2) -> 2 accumulators, A reused.
// ---------------------------------------------------------------------------
__global__ void __launch_bounds__(256)
shiftlsa_stageB(const float* __restrict__ F1t, const float* __restrict__ F2t,
                float* __restrict__ G) {
    __shared__ float As[64 * 68];
    __shared__ float Bs[64 * 68];

    const int b  = blockIdx.z;
    const int n0 = blockIdx.x * 64;
    const int m0 = blockIdx.y * 64;

    const float* Fa = F1t + (size_t)b * NPIX * CDIM;
    const float* Fb = F2t + (size_t)b * NPIX * CDIM;

    const int lane = threadIdx.x & 31;
    const int w    = threadIdx.x >> 5;
    const int twr  = w & 3;        // tile row 0..3   (rows twr*16..)
    const int twc  = w >> 2;       // tile col pair 0..1 (cols twc*32..)

    v8f acc0 = {};
    v8f acc1 = {};

    const int arow  = twr * 16 + (lane & 15);
    const int bcol0 = twc * 32 + (lane & 15);
    const int bcol1 = bcol0 + 16;
    const int koff  = (lane >> 4) << 1;   // lanes 16..31 hold K+2 (ISA 7.12.2)

    for (int kb = 0; kb < CDIM; kb += 64) {
        // Cooperative tile load: 64 rows x 64 K-floats, float4 per thread x4.
#pragma unroll
        for (int it = 0; it < 4; ++it) {
            const int f   = threadIdx.x + 256 * it;  // 0..1023
            const int row = f >> 4;
            const int cc  = (f & 15) << 2;
            *(float4*)(&As[row * 68 + cc]) =
                *(const float4*)(Fa + ((size_t)(n0 + row)) * CDIM + kb + cc);
            *(float4*)(&Bs[row * 68 + cc]) =
                *(const float4*)(Fb + ((size_t)(m0 + row)) * CDIM + kb + cc);
        }
        __syncthreads();

#pragma unroll
        for (int k = 0; k < 64; k += 4) {
            const v2f a  = *(const v2f*)(&As[arow  * 68 + k + koff]);
            const v2f b0 = *(const v2f*)(&Bs[bcol0 * 68 + k + koff]);
            const v2f b1 = *(const v2f*)(&Bs[bcol1 * 68 + k + koff]);
            acc0 = __builtin_amdgcn_wmma_f32_16x16x4_f32(
                       false, a, false, b0, (short)0, acc0, false, false);
            acc1 = __builtin_amdgcn_wmma_f32_16x16x4_f32(
                       false, a, false, b1, (short)0, acc1, false, false);
        }
        __syncthreads();
    }

    // C/D layout: lane -> N = lane&15, M = r + 8*(lane>>4) for vgpr r.
    float* Gp = G + (size_t)b * NPIX * NPIX;
    const int orow  = n0 + twr * 16 + ((lane >> 4) << 3);
    const int ocol  = m0 + twc * 32 + (lane & 15);
#pragma unroll
    for (int r = 0; r < 8; ++r) {
        Gp[(size_t)(orow + r) * NPIX + ocol]      = acc0[r];
        Gp[(size_t)(orow + r) * NPIX + ocol + 16] = acc1[r];
    }
}

// ---------------------------------------------------------------------------
// Stage C: corr[b,n,m] = scale * sum_l attn[b,n,l] * G[b, p(n,l), p(m,l)].
// One block per (b, n): 25 attn scalars via LDS; thread t owns column
// X = t&63 and 16 Y rows -> coalesced L2-resident gathers of shifted G rows.
// ---------------------------------------------------------------------------
__global__ void __launch_bounds__(256)
shiftlsa_stageC(const float* __restrict__ attn, const float* __restrict__ G,
                float* __restrict__ out) {
    const int b = blockIdx.y;
    const int n = blockIdx.x;
    const int y = n >> 6;
    const int x = n & 63;

    __shared__ float sA[32];
    if (threadIdx.x < LWIN)
        sA[threadIdx.x] = attn[((size_t)b * NPIX + n) * LWIN + threadIdx.x];
    __syncthreads();

    const int X  = threadIdx.x & 63;
    const int yg = threadIdx.x >> 6;            // 0..3, owns Y = yg*16 + q

    float acc[16];
#pragma unroll
    for (int q = 0; q < 16; ++q) acc[q] = 0.f;

    const float* Gb = G + (size_t)b * NPIX * NPIX;

#pragma unroll
    for (int l = 0; l < LWIN; ++l) {
        const int i = l / 5 - 2;
        const int j = l % 5 - 2;
        const float a = sA[l];
        const int py = clamp6(y + i);
        const int px = clamp6(x + j);
        const float* Grow = Gb + ((size_t)(py * 64 + px)) * NPIX;
        const int cX = clamp6(X + j);
#pragma unroll
        for (int q = 0; q < 16; ++q) {
            const int cY = clamp6(yg * 16 + q + i);
            acc[q] = fmaf(a, Grow[cY * 64 + cX], acc[q]);
        }
    }

    const float scale = 0.0883883476483184405f;   // 1/sqrt(128)
    const size_t base = ((size_t)b * NPIX + n) * NPIX;
#pragma unroll
    for (int q = 0; q < 16; ++q)
        out[base + (size_t)(yg * 16 + q) * 64 + X] = acc[q] * scale;
}

// ---------------------------------------------------------------------------
extern "C" void kernel_launch(void* const* d_in, const int* in_sizes, int n_in,
                              void* d_out, int out_size, void* d_ws, size_t ws_size,
                              hipStream_t stream) {
    const float* attn  = (const float*)d_in[0];  // (2,1,4096,25)
    const float* fmap  = (const float*)d_in[1];  // (2,128,64,64)
    const float* fmap2 = (const float*)d_in[2];  // (2,128,64,64)
    const float* W1    = (const float*)d_in[3];  // (128,128)
    const float* W2    = (const float*)d_in[4];  // (128,128)
    float*       out   = (float*)d_out;          // (2,64,64,1,64,64)

    // Workspace carve-up: F1t (4 MB) + F2t (4 MB) + G (128 MB)
    float* F1t = (float*)d_ws;
    float* F2t = F1t + (size_t)2 * NPIX * CDIM;
    float* G   = F2t + (size_t)2 * NPIX * CDIM;

    // Stage A: 1x1 convs, pixel-major output.
    shiftlsa_stageA<<<dim3(4096), dim3(256), 0, stream>>>(
        fmap, fmap2, W1, W2, F1t, F2t);

    // Stage B: Gram matrix via fp32 WMMA, 64x64 tiles, per batch.
    shiftlsa_stageB<<<dim3(64, 64, 2), dim3(256), 0, stream>>>(F1t, F2t, G);

    // Stage C: attn-weighted diagonal-shift accumulation over G.
    shiftlsa_stageC<<<dim3(4096, 2), dim3(256), 0, stream>>>(attn, G, out);
}